// SignedGCN_37675453120777
// MI455X (gfx1250) — compile-verified
//
#include <hip/hip_runtime.h>

typedef float v2f __attribute__((ext_vector_type(2)));
typedef float v8f __attribute__((ext_vector_type(8)));

#define NS0 300000
#define ND0 50000
#define NE0 800000
#define NS1 50000
#define ND1 8192
#define NE1 131072
#define INF 256
#define HIDF 128
#define OUTF 128

__device__ __forceinline__ float inv_sqrt_deg(unsigned d) {
    unsigned c = d ? d : 1u;            // clamp degree >= 1
    return rsqrtf((float)c);
}

// ---------------- workspace zeroing ----------------
__global__ void zero_ws_kernel(float* __restrict__ p, size_t n) {
    size_t i = (size_t)blockIdx.x * blockDim.x + threadIdx.x;
    size_t stride = (size_t)gridDim.x * blockDim.x;
    for (; i < n; i += stride) p[i] = 0.0f;
}

// ---------------- unweighted degree counting ----------------
__global__ void degree_kernel(const int* __restrict__ src, const int* __restrict__ dst,
                              int ne, unsigned* __restrict__ dsrc, unsigned* __restrict__ ddst) {
    int e = blockIdx.x * blockDim.x + threadIdx.x;
    if (e < ne) {
        atomicAdd(&dsrc[src[e]], 1u);
        atomicAdd(&ddst[dst[e]], 1u);
    }
}

// ---------------- edge scatter, 256 features, one wave per edge ----------------
__global__ void scatter256_kernel(const float* __restrict__ x, const int* __restrict__ src,
                                  const int* __restrict__ dst, const float* __restrict__ ew,
                                  const unsigned* __restrict__ deg_out, float* __restrict__ agg) {
    int w = blockIdx.x * (blockDim.x >> 5) + (threadIdx.x >> 5);
    if (w >= NE0) return;
    int lane = threadIdx.x & 31;
    int s = src[w], d = dst[w];
    float wt = ew[w] * inv_sqrt_deg(deg_out[s]);
    const float4* xr = (const float4*)(x + (size_t)s * INF);
    float* ar = agg + (size_t)d * INF;
    float4 a0 = xr[lane];
    float4 a1 = xr[lane + 32];
    int b0 = lane * 4, b1 = (lane + 32) * 4;
    atomicAdd(&ar[b0 + 0], wt * a0.x);
    atomicAdd(&ar[b0 + 1], wt * a0.y);
    atomicAdd(&ar[b0 + 2], wt * a0.z);
    atomicAdd(&ar[b0 + 3], wt * a0.w);
    atomicAdd(&ar[b1 + 0], wt * a1.x);
    atomicAdd(&ar[b1 + 1], wt * a1.y);
    atomicAdd(&ar[b1 + 2], wt * a1.z);
    atomicAdd(&ar[b1 + 3], wt * a1.w);
}

// ---------------- edge scatter, 128 features, one wave per edge ----------------
__global__ void scatter128_kernel(const float* __restrict__ h, const int* __restrict__ src,
                                  const int* __restrict__ dst, const float* __restrict__ ew,
                                  float* __restrict__ agg) {
    int w = blockIdx.x * (blockDim.x >> 5) + (threadIdx.x >> 5);
    if (w >= NE1) return;
    int lane = threadIdx.x & 31;
    int s = src[w], d = dst[w];
    float wt = ew[w];                   // source rows were pre-scaled in gemm1 epilogue
    const float4* hr = (const float4*)(h + (size_t)s * HIDF);
    float* ar = agg + (size_t)d * HIDF;
    float4 a0 = hr[lane];
    int b0 = lane * 4;
    atomicAdd(&ar[b0 + 0], wt * a0.x);
    atomicAdd(&ar[b0 + 1], wt * a0.y);
    atomicAdd(&ar[b0 + 2], wt * a0.z);
    atomicAdd(&ar[b0 + 3], wt * a0.w);
}

// ---------------- FP32 WMMA GEMM: out = relu(rs_in[m]*A[m,:] @ W + bias) * rs_out[m] ----------------
// A: [M x K] (K = 256 or 128), W: [K x 128], out: [M x 128].
// One wave computes a 16(M) x 128(N) strip: 8 accumulators of V_WMMA_F32_16X16X4_F32.
template <int K>
__global__ void gemm_wmma_kernel(const float* __restrict__ A, const float* __restrict__ W,
                                 const float* __restrict__ bias,
                                 const unsigned* __restrict__ deg_in,         // row scale (pre-GEMM)
                                 const unsigned* __restrict__ deg_out_next,   // row scale (post-ReLU), may be null
                                 float* __restrict__ out, int mtiles) {
    int wave = blockIdx.x * 4 + (threadIdx.x >> 5);
    if (wave >= mtiles) return;
    int lane = threadIdx.x & 31;
    int r15 = lane & 15;
    int khalf = (lane >> 4) << 1;        // 0 for lanes 0-15, 2 for lanes 16-31
    int m0 = wave * 16;
    int arow = m0 + r15;
    float rs = inv_sqrt_deg(deg_in[arow]);
    const float* Ar = A + (size_t)arow * K;

    v8f acc[8] = {};
    for (int k0 = 0; k0 < K; k0 += 4) {
        // A fragment: 16x4 f32 (M = lane&15; K = k0 + khalf + {0,1})
        v2f a;
        a.x = Ar[k0 + khalf + 0] * rs;
        a.y = Ar[k0 + khalf + 1] * rs;
        const float* Wk = W + (size_t)(k0 + khalf) * 128;
#pragma unroll
        for (int nt = 0; nt < 8; ++nt) {
            // B fragment: 4x16 f32 (N = nt*16 + (lane&15); K = k0 + khalf + {0,1})
            int col = nt * 16 + r15;
            v2f b;
            b.x = Wk[col];
            b.y = Wk[128 + col];
            acc[nt] = __builtin_amdgcn_wmma_f32_16x16x4_f32(
                false, a, false, b, (short)0, acc[nt], false, false);
        }
    }

    // Epilogue: D layout -> VGPR r holds row m0 + r (lanes 0-15) / m0 + 8 + r (lanes 16-31)
    int rowhalf = (lane >> 4) * 8;
#pragma unroll
    for (int r = 0; r < 8; ++r) {
        int row = m0 + rowhalf + r;
        float osr = deg_out_next ? inv_sqrt_deg(deg_out_next[row]) : 1.0f;
#pragma unroll
        for (int nt = 0; nt < 8; ++nt) {
            int col = nt * 16 + r15;
            float v = acc[nt][r] + bias[col];
            v = v > 0.0f ? v : 0.0f;
            out[(size_t)row * 128 + col] = v * osr;
        }
    }
}

extern "C" void kernel_launch(void* const* d_in, const int* in_sizes, int n_in,
                              void* d_out, int out_size, void* d_ws, size_t ws_size,
                              hipStream_t stream) {
    const float* x    = (const float*)d_in[0];
    const int*   src0 = (const int*)d_in[1];
    const int*   dst0 = (const int*)d_in[2];
    const float* ew0  = (const float*)d_in[3];
    const int*   src1 = (const int*)d_in[4];
    const int*   dst1 = (const int*)d_in[5];
    const float* ew1  = (const float*)d_in[6];
    const float* W1   = (const float*)d_in[7];
    const float* b1   = (const float*)d_in[8];
    const float* W2   = (const float*)d_in[9];
    const float* b2   = (const float*)d_in[10];
    float* out = (float*)d_out;

    // Workspace layout (4-byte elems), ~83 MB total:
    float*    agg0     = (float*)d_ws;                       // ND0 x 256
    float*    hs       = agg0 + (size_t)ND0 * INF;           // ND0 x 128 (pre-scaled layer-2 input)
    float*    agg1     = hs + (size_t)ND0 * HIDF;            // ND1 x 128
    unsigned* deg_out0 = (unsigned*)(agg1 + (size_t)ND1 * HIDF);
    unsigned* deg_in0  = deg_out0 + NS0;
    unsigned* deg_out1 = deg_in0 + ND0;
    unsigned* deg_in1  = deg_out1 + NS1;
    size_t total_elems = (size_t)((unsigned*)deg_in1 + ND1 - (unsigned*)d_ws);

    // 1) zero accumulators + degree counters (hs is fully overwritten later)
    zero_ws_kernel<<<2048, 256, 0, stream>>>((float*)d_ws, total_elems);

    // 2) unweighted degrees for both blocks
    degree_kernel<<<(NE0 + 255) / 256, 256, 0, stream>>>(src0, dst0, NE0, deg_out0, deg_in0);
    degree_kernel<<<(NE1 + 255) / 256, 256, 0, stream>>>(src1, dst1, NE1, deg_out1, deg_in1);

    // 3) layer-1 aggregation: agg0[dst] += ew * x[src] * outdeg0^-1/2   (one wave per edge)
    scatter256_kernel<<<NE0 / 8, 256, 0, stream>>>(x, src0, dst0, ew0, deg_out0, agg0);

    // 4) layer-1 GEMM + fused epilogue:
    //    hs = relu(indeg0^-1/2 * (agg0 @ W1) + b1) * outdeg1^-1/2
    gemm_wmma_kernel<INF><<<(ND0 / 16 + 3) / 4, 128, 0, stream>>>(
        agg0, W1, b1, deg_in0, deg_out1, hs, ND0 / 16);

    // 5) layer-2 aggregation: agg1[dst] += ew * hs[src]   (one wave per edge)
    scatter128_kernel<<<NE1 / 8, 256, 0, stream>>>(hs, src1, dst1, ew1, agg1);

    // 6) layer-2 GEMM + fused epilogue: out = relu(indeg1^-1/2 * (agg1 @ W2) + b2)
    gemm_wmma_kernel<HIDF><<<(ND1 / 16) / 4, 128, 0, stream>>>(
        agg1, W2, b2, deg_in1, nullptr, out, ND1 / 16);
}